// DemandRouter_60687887892694
// MI455X (gfx1250) — compile-verified
//
#include <hip/hip_runtime.h>
#include <hip/hip_bf16.h>
#include <cfloat>

// ---------------------------------------------------------------------------
// DemandRouter for MI455X (gfx1250, wave32)
//   x  : [4, 2048, 1024] f32
//   Wq : [32, 1024], bq : [32], Wk : [32, 1024], bk : [32]
//   out: [4, 2048, 1024] f32 = mean over top-4 (by q.k) neighbor rows of x
// ---------------------------------------------------------------------------

#define B   4
#define T   2048
#define D   1024
#define KQ  32
#define NROWS (B * T)          // 8192

typedef float v2f __attribute__((ext_vector_type(2)));
typedef float v8f __attribute__((ext_vector_type(8)));

// ---------------------------------------------------------------------------
// Kernel 1: Q/K projection via V_WMMA_F32_16X16X4_F32.
// Logical GEMM: C[8192 x 64] = X[8192 x 1024] * W^T, W = [Wq; Wk] (64 x 1024).
// One wave computes one 16x16 tile of C. Block = 128 threads = 4 waves,
// wave w handles the w-th 16-column tile (w<2 -> Wq cols, w>=2 -> Wk cols).
//
// fp32 WMMA fragment layout (ISA 7.12.2, 32-bit A 16x4 / B 4x16):
//   lane l, mi = l&15, hi = 2*(l>>4):
//     A vgprs = { X[row0+mi, d0+hi], X[row0+mi, d0+hi+1] }
//     B vgprs = { W[n0 +mi, d0+hi], W[n0 +mi, d0+hi+1] }   (B[k,n] = W[n, d0+k])
//   D vgpr r, lane l: element (m = r + 8*(l>>4), n = l&15).
// ---------------------------------------------------------------------------
__global__ void __launch_bounds__(128)
dr_proj_wmma(const float* __restrict__ x,
             const float* __restrict__ Wq, const float* __restrict__ bq,
             const float* __restrict__ Wk, const float* __restrict__ bk,
             float* __restrict__ Qout, float* __restrict__ Kout)
{
    const int lane = threadIdx.x & 31;
    const int wv   = threadIdx.x >> 5;          // 0..3 : which 16-col tile
    const int mi   = lane & 15;
    const int hi   = (lane >> 4) << 1;          // 0 or 2
    const int row0 = blockIdx.x << 4;           // 16 rows per block

    const bool isK  = (wv >= 2);
    const int  nc0  = (wv & 1) << 4;            // col offset inside Wq/Wk: 0|16
    const float* W    = isK ? Wk : Wq;
    const float* bias = isK ? bk : bq;

    const float* xrow = x + (size_t)(row0 + mi) * D + hi;
    const float* wrow = W + (size_t)(nc0  + mi) * D + hi;

    v8f acc = {};
#pragma unroll 8
    for (int d0 = 0; d0 < D; d0 += 4) {
        v2f a = *(const v2f*)(xrow + d0);
        v2f b = *(const v2f*)(wrow + d0);
        acc = __builtin_amdgcn_wmma_f32_16x16x4_f32(
            /*neg_a=*/false, a, /*neg_b=*/false, b,
            /*c_mod=*/(short)0, acc, /*reuse_a=*/false, /*reuse_b=*/false);
    }

    const int   nn    = nc0 + mi;                   // column within Q/K (0..31)
    const float bv    = bias[nn];
    float*      obase = isK ? Kout : Qout;
    const int   mb    = row0 + ((lane >> 4) << 3);  // m = mb + r
#pragma unroll
    for (int r = 0; r < 8; ++r)
        obase[(size_t)(mb + r) * KQ + nn] = acc[r] + bv;
}

// ---------------------------------------------------------------------------
// Kernel 2: fused similarity + top-4 (sim never hits memory).
// Block = 128 threads, one query row t per thread. K[b] streamed through LDS
// in 256-row chunks (32 KB); all lanes scan the same s -> LDS broadcast.
// Strict '>' keeps the earliest index on ties (jax top_k tie-break).
// Positive scale 1/sqrt(32) dropped: it cannot change top-k order.
// ---------------------------------------------------------------------------
#define CHUNK 256
__global__ void __launch_bounds__(128)
dr_topk(const float* __restrict__ Q, const float* __restrict__ Kmat,
        int* __restrict__ topk)
{
    __shared__ float lds[CHUNK * KQ];              // 32 KB

    const int b = blockIdx.y;
    const int t = blockIdx.x * 128 + threadIdx.x;

    float q[KQ];
    {
        const float* qp = Q + ((size_t)b * T + t) * KQ;
#pragma unroll
        for (int j = 0; j < KQ; ++j) q[j] = qp[j];
    }

    float v0 = -FLT_MAX, v1 = -FLT_MAX, v2 = -FLT_MAX, v3 = -FLT_MAX;
    int   i0 = 0, i1 = 0, i2 = 0, i3 = 0;

    const float* kb = Kmat + (size_t)b * T * KQ;

    for (int c = 0; c < T; c += CHUNK) {
        __syncthreads();
        // cooperative coalesced load: CHUNK*KQ floats = 2048 float4
        const float4* src = (const float4*)(kb + (size_t)c * KQ);
        float4*       dst = (float4*)lds;
        for (int i = threadIdx.x; i < CHUNK * KQ / 4; i += 128) dst[i] = src[i];
        __syncthreads();

        for (int s = 0; s < CHUNK; ++s) {
            const float* kv = lds + s * KQ;
            float d = 0.f;
#pragma unroll
            for (int j = 0; j < KQ; ++j) d += q[j] * kv[j];
            const int si = c + s;
            if (d > v3) {
                if (d > v0)      { v3=v2; i3=i2; v2=v1; i2=i1; v1=v0; i1=i0; v0=d; i0=si; }
                else if (d > v1) { v3=v2; i3=i2; v2=v1; i2=i1; v1=d;  i1=si; }
                else if (d > v2) { v3=v2; i3=i2; v2=d;  i2=si; }
                else             { v3=d;  i3=si; }
            }
        }
    }

    int4 r; r.x = i0; r.y = i1; r.z = i2; r.w = i3;
    ((int4*)topk)[(size_t)b * T + t] = r;
}

// ---------------------------------------------------------------------------
// Kernel 3: gather-mean. One block per (b,t); 256 threads x float4 = 1024
// floats. Neighbor rows are 4 KB contiguous -> coalesced b128 loads; x fits
// in the 192 MB L2, so the 4x re-read stays on-chip.
// ---------------------------------------------------------------------------
__global__ void __launch_bounds__(256)
dr_gather(const float* __restrict__ x, const int* __restrict__ topk,
          float* __restrict__ out)
{
    const int bt = blockIdx.x;                 // 0..8191
    const int b  = bt >> 11;                   // /T
    const int4 idx = ((const int4*)topk)[bt];
    const float* xb = x + (size_t)b * T * D;
    const int d = threadIdx.x << 2;

    float4 a0 = *(const float4*)(xb + (size_t)idx.x * D + d);
    float4 a1 = *(const float4*)(xb + (size_t)idx.y * D + d);
    float4 a2 = *(const float4*)(xb + (size_t)idx.z * D + d);
    float4 a3 = *(const float4*)(xb + (size_t)idx.w * D + d);

    float4 r;
    r.x = (a0.x + a1.x + a2.x + a3.x) * 0.25f;
    r.y = (a0.y + a1.y + a2.y + a3.y) * 0.25f;
    r.z = (a0.z + a1.z + a2.z + a3.z) * 0.25f;
    r.w = (a0.w + a1.w + a2.w + a3.w) * 0.25f;

    *(float4*)(out + (size_t)bt * D + d) = r;
}

// ---------------------------------------------------------------------------
extern "C" void kernel_launch(void* const* d_in, const int* in_sizes, int n_in,
                              void* d_out, int out_size, void* d_ws, size_t ws_size,
                              hipStream_t stream) {
    const float* x  = (const float*)d_in[0];
    const float* Wq = (const float*)d_in[1];
    const float* bq = (const float*)d_in[2];
    const float* Wk = (const float*)d_in[3];
    const float* bk = (const float*)d_in[4];
    float* out = (float*)d_out;

    // workspace: Q [8192*32] f32 | K [8192*32] f32 | idx [8192*4] i32
    float* Qw   = (float*)d_ws;
    float* Kw   = Qw + (size_t)NROWS * KQ;
    int*   idxw = (int*)(Kw + (size_t)NROWS * KQ);

    dr_proj_wmma<<<NROWS / 16, 128, 0, stream>>>(x, Wq, bq, Wk, bk, Qw, Kw);

    dim3 g2(T / 128, B);
    dr_topk<<<g2, 128, 0, stream>>>(Qw, Kw, idxw);

    dr_gather<<<NROWS, 256, 0, stream>>>(x, idxw, out);
}